// Attention_8753143349712
// MI455X (gfx1250) — compile-verified
//
#include <hip/hip_runtime.h>
#include <hip/hip_bf16.h>
#include <math.h>

typedef __attribute__((ext_vector_type(2))) float v2f;
typedef __attribute__((ext_vector_type(4))) float v4f;
typedef __attribute__((ext_vector_type(8))) float v8f;

#define B_DIM 256
#define T_DIM 1024
#define E_DIM 512
#define F_DIM 128
#define C_DIM 32
#define KW    31

// ---------------------------------------------------------------------------
// Kernel 1: fold conv filter + channel projection into one combined filter.
//   Wc[k][f] = sum_c W_loc[f,c] * conv_w[c,0,k]   (k-major, k=31 row zeroed)
//   biasF[f] = sum_c W_loc[f,c] * conv_b[c]
// ---------------------------------------------------------------------------
__global__ __launch_bounds__(128) void prep_kernel(
    const float* __restrict__ Wloc,   // (128,32)
    const float* __restrict__ convw,  // (32,1,31)
    const float* __restrict__ convb,  // (32)
    float* __restrict__ Wc,           // (32,128) k-major
    float* __restrict__ biasF)        // (128)
{
    const int f = threadIdx.x;
    float wrow[C_DIM];
#pragma unroll
    for (int c = 0; c < C_DIM; ++c) wrow[c] = Wloc[f * C_DIM + c];

    float bias = 0.f;
#pragma unroll
    for (int c = 0; c < C_DIM; ++c) bias += wrow[c] * convb[c];
    biasF[f] = bias;

    for (int k = 0; k < KW; ++k) {
        float s = 0.f;
#pragma unroll
        for (int c = 0; c < C_DIM; ++c) s += wrow[c] * convw[c * KW + k];
        Wc[k * F_DIM + f] = s;
    }
    Wc[KW * F_DIM + f] = 0.f;   // zero pad row so K can be 32 for WMMA
}

// ---------------------------------------------------------------------------
// Kernel 2: processed_lstm P[b][f] = sum_k lstm[b][k] * W_lstm[f][k]
// M=256 (b), N=128 (f), K=1024 f32 GEMM via V_WMMA_F32_16X16X4_F32.
// One wave per 16x16 output tile (128 tiles).
// ---------------------------------------------------------------------------
__global__ __launch_bounds__(32) void plstm_wmma(
    const float* __restrict__ lstm,   // (256,1024)
    const float* __restrict__ Wl,     // (128,1024)
    float* __restrict__ P)            // (256,128)
{
    const int tile = blockIdx.x;            // 0..127
    const int f0 = (tile & 7) * 16;
    const int b0 = (tile >> 3) * 16;
    const int lane = threadIdx.x & 31;
    const int half = lane >> 4;             // 0: K lo pair, 1: K hi pair
    const int l16  = lane & 15;

    v8f c = {};
    const float* arow = lstm + (size_t)(b0 + l16) * 1024;
    const float* brow = Wl   + (size_t)(f0 + l16) * 1024;
#pragma unroll 8
    for (int k0 = 0; k0 < 1024; k0 += 4) {
        const int kA = k0 + half * 2;
        v2f a = *(const v2f*)(arow + kA);   // A[m=l16][kA..kA+1]
        v2f b = *(const v2f*)(brow + kA);   // B[kA..kA+1][n=l16] (= W_lstm[n][k])
        c = __builtin_amdgcn_wmma_f32_16x16x4_f32(
                false, a, false, b, (short)0, c, false, false);
    }
#pragma unroll
    for (int r = 0; r < 8; ++r)
        P[(size_t)(b0 + r + half * 8) * F_DIM + f0 + l16] = c[r];
}

// ---------------------------------------------------------------------------
// Kernel 3: fused per-batch-row attention.
//   Phase A: energies via WMMA GEMM (T x 32 x 128) + tanh + W_e reduction
//   Phase B: softmax over T in LDS
//   Phase C: context = w . encoder_output, NT b128 streaming loads
// Grid: 256 blocks (one per b) x 256 threads (8 waves).
// ---------------------------------------------------------------------------
__global__ __launch_bounds__(256) void attn_fused(
    const float* __restrict__ enc,    // (B,T,E)
    const float* __restrict__ pe,     // (B,T,F)
    const float* __restrict__ awc,    // (B,T)
    const float* __restrict__ Wc,     // (32,128) k-major
    const float* __restrict__ biasF,  // (128)
    const float* __restrict__ Pm,     // (B,128)
    const float* __restrict__ We,     // (128)
    float* __restrict__ out_ctx,      // (B,512)
    float* __restrict__ out_w)        // (B,1024)
{
    const int b    = blockIdx.x;
    const int tid  = threadIdx.x;
    const int lane = tid & 31;
    const int wave = tid >> 5;           // 0..7
    const int half = lane >> 4;
    const int l16  = lane & 15;

    __shared__ float s_x[T_DIM + 2 * (KW / 2) + 2]; // 1056: padded cum-weights
    __shared__ alignas(16) float s_wc[32 * F_DIM];  // 16 KB filter / ctx scratch
    __shared__ float s_pf[F_DIM];                   // P[b][f] + biasF[f]
    __shared__ float s_we[F_DIM];
    __shared__ float s_e[T_DIM];                    // energies -> weights
    __shared__ float s_red[16];

    // ---- stage LDS ----
    for (int i = tid; i < T_DIM + 32; i += 256) {
        float v = 0.f;
        int src = i - 15;
        if (src >= 0 && src < T_DIM) v = awc[(size_t)b * T_DIM + src];
        s_x[i] = v;
    }
    for (int i = tid; i < 32 * F_DIM; i += 256) s_wc[i] = Wc[i];
    if (tid < F_DIM) {
        s_pf[tid] = Pm[(size_t)b * F_DIM + tid] + biasF[tid];
        s_we[tid] = We[tid];
    }
    __syncthreads();

    // ---- Phase A: energies ----
    const float* perow = pe + (size_t)b * T_DIM * F_DIM;
    for (int tt = wave; tt < T_DIM / 16; tt += 8) {
        const int t0 = tt * 16;
        float eacc[8];
#pragma unroll
        for (int r = 0; r < 8; ++r) eacc[r] = 0.f;

        for (int ft = 0; ft < F_DIM / 16; ++ft) {
            const int f0 = ft * 16;
            v8f c = {};
#pragma unroll
            for (int k0 = 0; k0 < 32; k0 += 4) {
                const int kA = k0 + half * 2;
                v2f a, bf;
                a.x  = s_x[t0 + l16 + kA];          // X[t][k] = xpad[t+k]
                a.y  = s_x[t0 + l16 + kA + 1];
                bf.x = s_wc[kA * F_DIM + f0 + l16]; // Wc[k][f]
                bf.y = s_wc[(kA + 1) * F_DIM + f0 + l16];
                c = __builtin_amdgcn_wmma_f32_16x16x4_f32(
                        false, a, false, bf, (short)0, c, false, false);
            }
            const int   f   = f0 + l16;
            const float pf  = s_pf[f];
            const float wef = s_we[f];
#pragma unroll
            for (int r = 0; r < 8; ++r) {
                const int t = t0 + r + half * 8;    // C/D row mapping
                // pe is streamed exactly once -> non-temporal
                float pev = __builtin_nontemporal_load(
                                &perow[(size_t)t * F_DIM + f]);
                float v = tanhf(c[r] + pf + pev);
                eacc[r] += v * wef;
            }
        }
        // reduce across the 16 lanes of each half-wave
#pragma unroll
        for (int r = 0; r < 8; ++r) {
            float v = eacc[r];
            v += __shfl_xor(v, 1, 32);
            v += __shfl_xor(v, 2, 32);
            v += __shfl_xor(v, 4, 32);
            v += __shfl_xor(v, 8, 32);
            if (l16 == 0) s_e[t0 + r + half * 8] = v;
        }
    }
    __syncthreads();

    // ---- Phase B: softmax over T ----
    float m = -INFINITY;
    for (int i = tid; i < T_DIM; i += 256) m = fmaxf(m, s_e[i]);
#pragma unroll
    for (int o = 16; o; o >>= 1) m = fmaxf(m, __shfl_xor(m, o, 32));
    if (lane == 0) s_red[wave] = m;
    __syncthreads();
    if (tid == 0) {
        float mm = s_red[0];
        for (int i = 1; i < 8; ++i) mm = fmaxf(mm, s_red[i]);
        s_red[0] = mm;
    }
    __syncthreads();
    m = s_red[0];

    float sum = 0.f;
    for (int i = tid; i < T_DIM; i += 256) {
        float ex = __expf(s_e[i] - m);
        s_e[i] = ex;
        sum += ex;
    }
#pragma unroll
    for (int o = 16; o; o >>= 1) sum += __shfl_xor(sum, o, 32);
    __syncthreads();                         // everyone done reading s_red[0]
    if (lane == 0) s_red[wave] = sum;
    __syncthreads();
    if (tid == 0) {
        float s2 = 0.f;
        for (int i = 0; i < 8; ++i) s2 += s_red[i];
        s_red[0] = s2;
    }
    __syncthreads();
    const float inv = 1.f / s_red[0];
    for (int i = tid; i < T_DIM; i += 256) {
        float w = s_e[i] * inv;
        s_e[i] = w;
        __builtin_nontemporal_store(w, &out_w[(size_t)b * T_DIM + i]);
    }
    __syncthreads();

    // ---- Phase C: context = sum_t w[t] * enc[b,t,:] ----
    // b128 NT loads: 128 float4 per row; threads split even/odd t rows.
    const v4f* enc4 = (const v4f*)(enc + (size_t)b * T_DIM * E_DIM);
    const int j4   = tid & 127;          // float4 column 0..127
    const int tsub = tid >> 7;           // 0: even rows, 1: odd rows
    v4f acc = {};
#pragma unroll 4
    for (int t = tsub; t < T_DIM; t += 2) {
        const float w = s_e[t];
        v4f v = __builtin_nontemporal_load(&enc4[(size_t)t * (E_DIM / 4) + j4]);
        acc += w * v;
    }
    // combine the two t-phases through LDS (reuse s_wc as scratch)
    v4f* s_ctx = (v4f*)s_wc;
    if (tsub == 0) s_ctx[j4] = acc;
    __syncthreads();
    if (tsub == 1) {
        v4f o = s_ctx[j4] + acc;
        __builtin_nontemporal_store(o, &((v4f*)(out_ctx + (size_t)b * E_DIM))[j4]);
    }
}

// ---------------------------------------------------------------------------
extern "C" void kernel_launch(void* const* d_in, const int* in_sizes, int n_in,
                              void* d_out, int out_size, void* d_ws, size_t ws_size,
                              hipStream_t stream) {
    const float* enc   = (const float*)d_in[0];   // (256,1024,512)
    const float* pe    = (const float*)d_in[1];   // (256,1024,128)
    const float* lstm  = (const float*)d_in[2];   // (256,1024)
    const float* awc   = (const float*)d_in[3];   // (256,1024)
    const float* Wl    = (const float*)d_in[4];   // (128,1024)
    const float* convw = (const float*)d_in[5];   // (32,1,31)
    const float* convb = (const float*)d_in[6];   // (32)
    const float* Wloc  = (const float*)d_in[7];   // (128,32)
    const float* We    = (const float*)d_in[8];   // (1,128)

    float* out_ctx = (float*)d_out;                       // (256,512)
    float* out_w   = out_ctx + (size_t)B_DIM * E_DIM;     // (256,1024)

    float* ws    = (float*)d_ws;
    float* Wc    = ws;                         // 32*128 floats
    float* biasF = ws + 32 * F_DIM;            // 128 floats
    float* P     = biasF + F_DIM;              // 256*128 floats

    prep_kernel<<<1, 128, 0, stream>>>(Wloc, convw, convb, Wc, biasF);
    plstm_wmma<<<128, 32, 0, stream>>>(lstm, Wl, P);
    attn_fused<<<B_DIM, 256, 0, stream>>>(enc, pe, awc, Wc, biasF, P, We,
                                          out_ctx, out_w);
}